// QueryAndGroupQuat_10574209482756
// MI455X (gfx1250) — compile-verified
//
#include <hip/hip_runtime.h>
#include <math.h>

// Problem constants (match reference setup_inputs)
#define NPTS   8192
#define NQ     2048
#define NB     4
#define QPB    16            // queries per wave/block
#define RADIUS 0.2f
#define R2     (RADIUS * RADIUS)
#define EPSF   1e-6f

typedef float v2f __attribute__((ext_vector_type(2)));
typedef float v8f __attribute__((ext_vector_type(8)));

// One wave32 per block. The wave owns 16 query points:
//   Stage 1: ball query via V_WMMA_F32_16X16X4_F32 Gram tiles (2 per iter),
//            ballots distributed by cndmask chain, parallel ctz-append.
//   Stage 2: per query, 32 lanes (= samples) do projection/angle/stable-sort
//            (rank + ds_permute scatter) and write all 34 output channels.
__global__ __launch_bounds__(32)
void qgq_kernel(const float* __restrict__ xyz,
                const float* __restrict__ new_xyz,
                float* __restrict__ out)
{
    __shared__ int s_idx[QPB][32];

    const int lane = threadIdx.x;             // 0..31
    const int blk  = blockIdx.x;              // 0..NB*(NQ/QPB)-1
    const int b    = blk / (NQ / QPB);
    const int q0   = (blk % (NQ / QPB)) * QPB;

    const float* __restrict__ xyzb = xyz     + (size_t)b * NPTS * 3;
    const float* __restrict__ ctrb = new_xyz + (size_t)b * NQ   * 3;

    // ---- Stage 1: ball query ------------------------------------------------
    // Lane l holds center of query (l & 15) (both halves keep a copy).
    const int mq = lane & 15;
    const float cx = ctrb[(q0 + mq) * 3 + 0];
    const float cy = ctrb[(q0 + mq) * 3 + 1];
    const float cz = ctrb[(q0 + mq) * 3 + 2];
    const float c2 = cx * cx + cy * cy + cz * cz;

    // A fragment: 16x4 f32 (M=query, K = {x,y,z,0}).
    // ISA layout: lanes 0-15 -> {K0,K1}, lanes 16-31 -> {K2,K3}.
    v2f A;
    if (lane < 16) { A.x = cx; A.y = cy; }
    else           { A.x = cz; A.y = 0.0f; }

    // |c_M|^2 for the 8 C rows this lane touches: M = v + 8*(lane>=16).
    float cm2[8];
    const int hi8 = (lane >= 16) ? 8 : 0;
#pragma unroll
    for (int v = 0; v < 8; ++v) cm2[v] = __shfl(c2, v + hi8, 32);

    const int lv   = lane & 7;         // which ballot this lane consumes
    int cnt = (lane < 16) ? 0 : 32;    // non-owner lanes report "full"

    for (int j0 = 0; j0 < NPTS; j0 += 32) {
        __builtin_prefetch(&xyzb[((j0 + 256) & (NPTS - 1)) * 3], 0, 0);

        // Two candidate half-tiles: n0 = j0 + (lane&15), n1 = n0 + 16.
        const int n0 = j0 + (lane & 15);
        const int n1 = n0 + 16;
        const float p0x = xyzb[n0 * 3 + 0];
        const float p0y = xyzb[n0 * 3 + 1];
        const float p0z = xyzb[n0 * 3 + 2];
        const float p1x = xyzb[n1 * 3 + 0];
        const float p1y = xyzb[n1 * 3 + 1];
        const float p1z = xyzb[n1 * 3 + 2];
        const float p20 = p0x * p0x + p0y * p0y + p0z * p0z;
        const float p21 = p1x * p1x + p1y * p1y + p1z * p1z;

        // B fragments: 4x16 f32 (K=component, N=candidate), mirrored layout.
        v2f B0, B1;
        if (lane < 16) { B0.x = p0x; B0.y = p0y; B1.x = p1x; B1.y = p1y; }
        else           { B0.x = p0z; B0.y = 0.f; B1.x = p1z; B1.y = 0.f; }

        v8f acc0 = {}, acc1 = {};
        acc0 = __builtin_amdgcn_wmma_f32_16x16x4_f32(
                   false, A, false, B0, (short)0, acc0, false, false);
        acc1 = __builtin_amdgcn_wmma_f32_16x16x4_f32(
                   false, A, false, B1, (short)0, acc1, false, false);

        // 8 wave-uniform ballots per half-tile; distribute so that owner lane m
        // ends up holding ballot (m & 7) in a VGPR (cndmask chain, SGPR sources).
        unsigned myb0 = 0u, myb1 = 0u;
#pragma unroll
        for (int v = 0; v < 8; ++v) {
            const float d20 = cm2[v] + p20 - 2.0f * acc0[v];
            const float d21 = cm2[v] + p21 - 2.0f * acc1[v];
            const unsigned bal0 = (unsigned)__ballot(d20 < R2);
            const unsigned bal1 = (unsigned)__ballot(d21 < R2);
            myb0 = (lv == v) ? bal0 : myb0;
            myb1 = (lv == v) ? bal1 : myb1;
        }

        // Owner lane m (0..15): query m; low 16 bits = queries 0..7 (M=v),
        // high 16 bits = queries 8..15 (M=v+8). Bit position = candidate n.
        unsigned bits0, bits1;
        if (lane < 8)       { bits0 = myb0 & 0xFFFFu; bits1 = myb1 & 0xFFFFu; }
        else if (lane < 16) { bits0 = myb0 >> 16;     bits1 = myb1 >> 16;     }
        else                { bits0 = 0u;             bits1 = 0u;             }

        // Parallel append across all 16 owner lanes, ascending index order.
        while (bits0 && cnt < 32) {
            const int t = __builtin_ctz(bits0);
            s_idx[lane][cnt] = j0 + t;
            ++cnt;
            bits0 &= bits0 - 1u;
        }
        while (bits1 && cnt < 32) {
            const int t = __builtin_ctz(bits1);
            s_idx[lane][cnt] = j0 + 16 + t;
            ++cnt;
            bits1 &= bits1 - 1u;
        }

        // Exact early exit: all 16 queries have 32 hits.
        if ((unsigned)__ballot(cnt >= 32) == 0xFFFFFFFFu) break;
    }

    __syncthreads();
    if (lane < 16) {
        const int first = (cnt > 0) ? s_idx[lane][0] : 0;   // reference padding rule
        for (int j = cnt; j < 32; ++j) s_idx[lane][j] = first;
    }
    __syncthreads();

    // ---- Stage 2: geometry, stable sort, quaternions ------------------------
    const float kPI = 3.14159265358979323846f;

    for (int q = 0; q < QPB; ++q) {
        const int p = q0 + q;
        const float ccx = __shfl(cx, q, 32);
        const float ccy = __shfl(cy, q, 32);
        const float ccz = __shfl(cz, q, 32);

        const int nidx = s_idx[q][lane];
        const float rx = xyzb[nidx * 3 + 0] - ccx;
        const float ry = xyzb[nidx * 3 + 1] - ccy;
        const float rz = xyzb[nidx * 3 + 2] - ccz;

        // u = normalize(center)
        const float cn  = sqrtf(ccx * ccx + ccy * ccy + ccz * ccz);
        const float icn = 1.0f / (cn + EPSF);
        const float ux = ccx * icn, uy = ccy * icn, uz = ccz * icn;

        // reference axis (_project_one)
        const bool colin = fabsf(ux) > (1.0f - 0.001f);
        float ex = colin ? (-uy * ux) : (1.0f - ux * ux);
        float ey = colin ? (1.0f - uy * uy) : (-ux * uy);
        float ez = colin ? (-uy * uz) : (-ux * uz);
        const float en  = sqrtf(ex * ex + ey * ey + ez * ez);
        const float ien = 1.0f / (en + EPSF);
        ex *= ien; ey *= ien; ez *= ien;

        // project rel onto plane perpendicular to u, normalize
        const float du = ux * rx + uy * ry + uz * rz;
        float qx = rx - du * ux;
        float qy = ry - du * uy;
        float qz = rz - du * uz;
        const float qn  = sqrtf(qx * qx + qy * qy + qz * qz);
        const float iqn = 1.0f / (qn + EPSF);
        qx *= iqn; qy *= iqn; qz *= iqn;

        // signed angle about u
        const float wx = ey * qz - ez * qy;
        const float wy = ez * qx - ex * qz;
        const float wz = ex * qy - ey * qx;
        const float sinus = wx * ux + wy * uy + wz * uz;
        const float cosng = ex * qx + ey * qy + ez * qz;
        float ang = atan2f(sinus, cosng);
        if (qx * qx + qy * qy + qz * qz < 1e-12f) {
            // degenerate: reference uses random tie-break; deterministic hash here
            unsigned h = (unsigned)(((b * NQ + p) << 5) | lane);
            h = h * 2654435761u + 0x9e3779b9u;
            h ^= h >> 16; h *= 0x85ebca6bu; h ^= h >> 13;
            ang = ((float)(h & 0xFFFFFFu) * (1.0f / 16777216.0f) - 0.5f) * (2.0f * kPI);
        }

        // stable rank over 32 lanes (matches stable argsort tiebreak)
        int rank = 0;
#pragma unroll 8
        for (int j = 0; j < 32; ++j) {
            const float aj = __shfl(ang, j, 32);
            rank += (int)((aj < ang) || (aj == ang && j < lane));
        }

        // forward permute (scatter): sorted[rank] = rel  -> ds_permute_b32
        const int dst = rank << 2;
        const float sx = __int_as_float(__builtin_amdgcn_ds_permute(dst, __float_as_int(rx)));
        const float sy = __int_as_float(__builtin_amdgcn_ds_permute(dst, __float_as_int(ry)));
        const float sz = __int_as_float(__builtin_amdgcn_ds_permute(dst, __float_as_int(rz)));

        // invariance channels
        const float rr    = sx * sx + sy * sy + sz * sz;
        const float norm2 = sqrtf(rr + 1e-12f);
        float cosang = (ux * sx + uy * sy + uz * sz) / norm2;
        cosang = fminf(fmaxf(cosang, -1.0f + EPSF), 1.0f - EPSF);
        const float invang = acosf(cosang);

        // quaternion lift
        const float dist  = sqrtf(rr);
        const float iod   = 1.0f / (dist + EPSF);
        const float theta = dist * (1.0f / RADIUS) * (kPI * 0.5f);
        float qv[4];
        qv[0] = cosf(theta);
        const float st = sinf(theta);
        qv[1] = st * sx * iod;
        qv[2] = st * sy * iod;
        qv[3] = st * sz * iod;

        // out[b][ch][p][s], ch: 0=norm, 1=angle, 2 + k*8 + m = q_k rolled by -m
        const size_t chstride = (size_t)NQ * 32;
        const size_t base = (((size_t)b * 34) * NQ + p) * 32;
        out[base + lane]            = norm2;
        out[base + chstride + lane] = invang;
#pragma unroll
        for (int k = 0; k < 4; ++k) {
#pragma unroll
            for (int mm = 0; mm < 8; ++mm) {
                const int ch = 2 + k * 8 + mm;
                const int s  = (lane - mm) & 31;   // roll(q,-m)[s] = q[(s+m)%32]
                out[base + (size_t)ch * chstride + s] = qv[k];
            }
        }
    }
}

extern "C" void kernel_launch(void* const* d_in, const int* in_sizes, int n_in,
                              void* d_out, int out_size, void* d_ws, size_t ws_size,
                              hipStream_t stream) {
    (void)in_sizes; (void)n_in; (void)out_size; (void)d_ws; (void)ws_size;
    const float* xyz     = (const float*)d_in[0];   // (B, N, 3) f32
    const float* new_xyz = (const float*)d_in[1];   // (B, P, 3) f32
    // d_in[2] (fps_idx) is dead: the fps column of the grouping is dropped
    // by grouped[..., 1:] in the reference.
    float* out = (float*)d_out;                     // (B, 34, P, 32) f32

    const int blocks = NB * (NQ / QPB);             // 4 * 128 = 512 waves
    qgq_kernel<<<blocks, 32, 0, stream>>>(xyz, new_xyz, out);
}